// ConvQuadInterp3d_47485158424866
// MI455X (gfx1250) — compile-verified
//
#include <hip/hip_runtime.h>
#include <stdint.h>

// ---------------------------------------------------------------------------
// ConvQuadInterp3d for MI455X (gfx1250): TDM-staged 27-point stencil.
// Memory-bound (~168 MB HBM traffic => ~7 us floor at 23.3 TB/s).
// CDNA5 path: tensor_load_to_lds (TDM DMA) + s_wait_tensorcnt staging,
// non-temporal output stores to keep the input tile traffic resident in L2.
// ---------------------------------------------------------------------------

typedef __attribute__((ext_vector_type(4))) unsigned int u32x4;
typedef __attribute__((ext_vector_type(8))) int          i32x8;
typedef __attribute__((ext_vector_type(4))) int          i32x4;

#define Bc 2
#define Dc 4
#define Hc 1024
#define Wc 1024
#define TH 8
#define TW 32
#define LH (TH + 2)
#define LW (TW + 2)
#define EPSC 1e-7f
#define BONUS 10.0f

__global__ __launch_bounds__(TW * TH) void ConvQuadInterp3d_kernel(
    const float* __restrict__ x, const float* __restrict__ nz,
    float* __restrict__ coords, float* __restrict__ ymax) {
  // Staged tile: all 4 depth planes, (TH+2) x (TW+2) halo window, z-major
  // contiguous layout matching the TDM's LDS fill order.
  __shared__ float tile[Dc][LH][LW];

  const int w0 = blockIdx.x * TW;
  const int h0 = blockIdx.y * TH;
  const int b  = blockIdx.z;

  // Clamp the staged-window start so the TDM tile is always fully in-range
  // (this reproduces edge-replicate padding via index clamping at read time).
  const int hs = min(max(h0 - 1, 0), Hc - LH);
  const int ws = min(max(w0 - 1, 0), Wc - LW);

  // --- Wave 0 issues one TDM 3-D tile load: (LW x LH x Dc) fp32 -> LDS ----
  if (threadIdx.y == 0) {
    const unsigned long long HW = (unsigned long long)Hc * (unsigned long long)Wc;
    const unsigned long long gaddr =
        (unsigned long long)(uintptr_t)x +
        4ull * (((unsigned long long)b * Dc * Hc + (unsigned long long)hs) * Wc +
                (unsigned long long)ws);
    const unsigned lds = (unsigned)(uintptr_t)(&tile[0][0][0]);

    // D# group 0: count=1 | lds_addr | global_addr[56:0] | type=2
    u32x4 g0;
    g0.x = 1u;
    g0.y = lds;
    g0.z = (unsigned)gaddr;
    g0.w = (unsigned)((gaddr >> 32) & 0x1FFFFFFull) | (2u << 30);

    // D# group 1: data_size=4B, tensor_dim0/1, tile_dim0/1/2, strides
    i32x8 g1;
    g1[0] = (2 << 16);                                   // data_size = 4 bytes
    g1[1] = (int)((Wc & 0xFFFF) << 16);                  // tensor_dim0 lo16
    g1[2] = (int)((Wc >> 16) | ((Hc & 0xFFFF) << 16));   // dim0 hi16 | dim1 lo16
    g1[3] = (int)((Hc >> 16) | (LW << 16));              // dim1 hi16 | tile_dim0
    g1[4] = (int)(LH | (Dc << 16));                      // tile_dim1 | tile_dim2
    g1[5] = (int)Wc;                                     // tensor_dim0_stride lo32
    g1[6] = (int)((unsigned)(HW & 0xFFFFull) << 16);     // strd0 hi16 | strd1 lo16
    g1[7] = (int)(HW >> 16);                             // tensor_dim1_stride[47:16]

    // D# group 2: tensor_dim2 = Dc (3-D tensor); dim3/tile_dim3 unused
    i32x4 g2;
    g2[0] = Dc;
    g2[1] = 0;
    g2[2] = (int)(HW * Dc);                              // dim2 stride (unused)
    g2[3] = 0;

    i32x4 g3 = {0, 0, 0, 0};

    // 6-arg clang-23 form: extra i32x8 slot (zero = terminating/null block).
    i32x8 g4 = {0, 0, 0, 0, 0, 0, 0, 0};

    __builtin_amdgcn_tensor_load_to_lds(g0, g1, g2, g3, g4, 0);
    __builtin_amdgcn_s_wait_tensorcnt((short)0);
  }
  __syncthreads();

  const int tx = threadIdx.x, ty = threadIdx.y;
  const int h = h0 + ty, w = w0 + tx;

  // Clamped local H/W indices (edge replicate), hoisted out of depth loop.
  int ly[3], lx[3];
#pragma unroll
  for (int k = 0; k < 3; ++k) {
    ly[k] = min(max(h + k - 1, 0), Hc - 1) - hs;
    lx[k] = min(max(w + k - 1, 0), Wc - 1) - ws;
  }

  // Broadcast 3x3 noise (uniform address -> SMEM / L2 hit).
  float n[9];
#pragma unroll
  for (int k = 0; k < 9; ++k) n[k] = nz[k] * EPSC;

  const size_t HWs = (size_t)Hc * Wc;
  const size_t plane = (size_t)h * Wc + w;
  const size_t cstride = (size_t)Dc * HWs;

#pragma unroll
  for (int d = 0; d < Dc; ++d) {
    int lz[3];
#pragma unroll
    for (int k = 0; k < 3; ++k) lz[k] = min(max(d + k - 1, 0), Dc - 1);

    // v[dd+1][dh+1][dw+1] == x[d+dd, h+dh, w+dw] with replicate clamp.
    float v[3][3][3];
#pragma unroll
    for (int zz = 0; zz < 3; ++zz)
#pragma unroll
      for (int yy = 0; yy < 3; ++yy)
#pragma unroll
        for (int xx = 0; xx < 3; ++xx)
          v[zz][yy][xx] = tile[lz[zz]][ly[yy]][lx[xx]];

    const float xc = v[1][1][1];

    // Gradients (kornia diff kernel; depth kernel flipped -> gz sign).
    const float b0 = 0.5f * (v[1][1][2] - v[1][1][0]);   // gx (W)
    const float b1 = 0.5f * (v[1][2][1] - v[1][0][1]);   // gy (H)
    const float b2 = 0.5f * (v[0][1][1] - v[2][1][1]);   // gz (D, flipped)

    // Second-order stencils.
    const float dxx = v[1][1][2] - 2.0f * xc + v[1][1][0];
    const float dyy = v[1][2][1] - 2.0f * xc + v[1][0][1];
    const float dss = v[2][1][1] - 2.0f * xc + v[0][1][1];
    const float dxy = 0.25f * (v[1][0][0] - v[1][0][2] - v[1][2][0] + v[1][2][2]);
    const float dys = 0.25f * (v[2][0][1] - v[2][2][1] - v[0][0][1] + v[0][2][1]);
    const float dxs = 0.25f * (v[2][1][0] - v[2][1][2] - v[0][1][0] + v[0][1][2]);

    // NMS: clamped 27-neighborhood max == -inf-padded maxpool (each clamped
    // sample duplicates a valid in-window sample, and the center is included).
    float m = v[0][0][0];
#pragma unroll
    for (int zz = 0; zz < 3; ++zz)
#pragma unroll
      for (int yy = 0; yy < 3; ++yy)
#pragma unroll
        for (int xx = 0; xx < 3; ++xx) m = fmaxf(m, v[zz][yy][xx]);
    const bool mask = (xc == m);

    // A = Hes + noise*eps (full 3x3, slightly non-symmetric).
    const float a00 = dxx + n[0], a01 = dxy + n[1], a02 = dxs + n[2];
    const float a10 = dxy + n[3], a11 = dyy + n[4], a12 = dys + n[5];
    const float a20 = dxs + n[6], a21 = dys + n[7], a22 = dss + n[8];

    // Cramer's rule solve A * s = b.
    const float c00 = a11 * a22 - a12 * a21;
    const float c10 = a10 * a22 - a12 * a20;
    const float c20 = a10 * a21 - a11 * a20;
    const float det = a00 * c00 - a01 * c10 + a02 * c20;
    const float rd = 1.0f / det;
    const float s0 = (b0 * c00 - a01 * (b1 * a22 - b2 * a12) +
                      a02 * (b1 * a21 - b2 * a11)) * rd;
    const float s1 = (a00 * (b1 * a22 - b2 * a12) - b0 * c10 +
                      a02 * (a10 * b2 - a20 * b1)) * rd;
    const float s2 = (a00 * (a11 * b2 - a21 * b1) -
                      a01 * (a10 * b2 - a20 * b1) + b0 * c20) * rd;

    float dx0 = mask ? -s0 : 0.0f;
    float dx1 = mask ? -s1 : 0.0f;
    float dx2 = mask ? -s2 : 0.0f;
    const float amax = fmaxf(fabsf(dx0), fmaxf(fabsf(dx1), fabsf(dx2)));
    if (amax > 0.7f) { dx0 = 0.0f; dx1 = 0.0f; dx2 = 0.0f; }

    const float dy = 0.5f * (b0 * dx0 + b1 * dx1 + b2 * dx2);
    const float y = xc + dy + (mask ? BONUS : 0.0f);

    // coords channel order is (d,h,w) = reversed (x,y,z) offsets.
    // Non-temporal: outputs are write-once; keep L2 for the stencil halos.
    float* cout = coords + (size_t)b * 3 * cstride + (size_t)d * HWs + plane;
    __builtin_nontemporal_store((float)d + dx2, cout);
    __builtin_nontemporal_store((float)h + dx1, cout + cstride);
    __builtin_nontemporal_store((float)w + dx0, cout + 2 * cstride);
    __builtin_nontemporal_store(y, ymax + ((size_t)b * Dc + d) * HWs + plane);
  }
}

extern "C" void kernel_launch(void* const* d_in, const int* in_sizes, int n_in,
                              void* d_out, int out_size, void* d_ws, size_t ws_size,
                              hipStream_t stream) {
  (void)in_sizes; (void)n_in; (void)out_size; (void)d_ws; (void)ws_size;
  const float* x  = (const float*)d_in[0];   // (2,1,4,1024,1024) fp32
  const float* nz = (const float*)d_in[1];   // (3,3) fp32
  float* out = (float*)d_out;
  float* coords = out;                                   // (2,1,3,4,1024,1024)
  float* ymax = out + (size_t)Bc * 3 * Dc * Hc * Wc;     // (2,1,4,1024,1024)

  dim3 block(TW, TH, 1);
  dim3 grid(Wc / TW, Hc / TH, Bc);
  ConvQuadInterp3d_kernel<<<grid, block, 0, stream>>>(x, nz, coords, ymax);
}